// FAE_13365938225175
// MI455X (gfx1250) — compile-verified
//
#include <hip/hip_runtime.h>

typedef float v2f __attribute__((ext_vector_type(2)));
typedef float v8f __attribute__((ext_vector_type(8)));
typedef unsigned int u32x4 __attribute__((ext_vector_type(4)));
typedef int i32x4 __attribute__((ext_vector_type(4)));
typedef int i32x8 __attribute__((ext_vector_type(8)));

#define BN_EPS 1e-5f

#if defined(__HIP_DEVICE_COMPILE__) && defined(__gfx1250__) && \
    __has_builtin(__builtin_amdgcn_tensor_load_to_lds) &&      \
    __has_builtin(__builtin_amdgcn_s_wait_tensorcnt)
#define USE_TDM 1
#else
#define USE_TDM 0
#endif

#if USE_TDM
// LDS byte offset of a generic pointer to a __shared__ object.
__device__ __forceinline__ unsigned lds_off(void* p) {
  return (unsigned)(size_t)(__attribute__((address_space(3))) void*)p;
}

// 1-D TDM copy: tile of `tile` f32 elements from gsrc -> LDS byte addr.
// tensor_dim0 = `valid` so reads past the logical end store zeros (tail pad).
// D# layout per CDNA5 ISA.
// This toolchain exposes the 6-arg builtin:
//   (u32x4 g0, i32x8 g1, i32x4 g2, i32x4 g3, i32x8 g4, i32 cpol)
__device__ __forceinline__ void tdm_load_1d(const float* gsrc, unsigned lds_byte,
                                            int valid, int tile)
{
  unsigned long long ga = (unsigned long long)(size_t)gsrc;
  u32x4 g0 = (u32x4)0u;
  g0.x = 1u;                                          // count=1, user mode
  g0.y = lds_byte;                                    // lds_addr [63:32]
  g0.z = (unsigned)(ga & 0xFFFFFFFFull);              // global_addr [95:64]
  g0.w = (unsigned)((ga >> 32) & 0x1FFFFFFull)        // global_addr [120:96]
         | 0x80000000u;                               // type=2 ("image") [127:126]
  i32x8 g1 = (i32x8)0;
  g1.s0 = 0x00020000;                                 // wg_mask=0, data_size=2 (4B)
  g1.s1 = (valid & 0xFFFF) << 16;                     // tensor_dim0 lo16 (bits 79:48)
  g1.s2 = ((valid >> 16) & 0xFFFF) | (1 << 16);       // tensor_dim0 hi | tensor_dim1=1
  g1.s3 = (tile & 0xFFFF) << 16;                      // tile_dim0 [127:112]
  g1.s4 = 1;                                          // tile_dim1=1, tile_dim2=0
  g1.s5 = tile;                                       // tensor_dim0_stride lo (unused, dim1=1)
  g1.s6 = 0;
  g1.s7 = 0;
  i32x4 z4 = (i32x4)0;
  i32x8 z8 = (i32x8)0;
  __builtin_amdgcn_tensor_load_to_lds(g0, g1, z4, z4, z8, 0);
}
#endif

// ---------------------------------------------------------------------------
// K1: encoder — one workgroup, LDS-resident intermediates.
// h = conv(x,e1); h = conv(relu(bn(h)),e2); h = conv(relu(bn(h)),e3);
// h = conv(bn(relu(h)),e4) -> H[800] (channel-major 32x25)
// ---------------------------------------------------------------------------
__global__ __launch_bounds__(1024) void k_encoder(
    const float* __restrict__ x,
    const float* __restrict__ e1w, const float* __restrict__ e1b,
    const float* __restrict__ bn2g, const float* __restrict__ bn2b,
    const float* __restrict__ e2w, const float* __restrict__ e2b,
    const float* __restrict__ bn3g, const float* __restrict__ bn3b,
    const float* __restrict__ e3w, const float* __restrict__ e3b,
    const float* __restrict__ bn4g, const float* __restrict__ bn4b,
    const float* __restrict__ e4w, const float* __restrict__ e4b,
    float* __restrict__ H)
{
  __shared__ float sA[256 * 31];
  __shared__ float sB[128 * 29];
  __shared__ float sC[64 * 27];
  __shared__ float ssc[256], ssh[256];
  const int tid = threadIdx.x;
  const int NT = 1024;

  // conv1: 1->256, L 33->31
  for (int i = tid; i < 256 * 31; i += NT) {
    int c = i / 31, t = i % 31;
    const float* w = e1w + c * 3;
    sA[i] = x[t] * w[0] + x[t + 1] * w[1] + x[t + 2] * w[2] + e1b[c];
  }
  __syncthreads();
  // bn2 stats
  if (tid < 256) {
    float s = 0.f, q = 0.f;
    for (int t = 0; t < 31; ++t) { float v = sA[tid * 31 + t]; s += v; q += v * v; }
    float mu = s * (1.f / 31.f);
    float var = q * (1.f / 31.f) - mu * mu;
    float sc = bn2g[tid] * rsqrtf(var + BN_EPS);
    ssc[tid] = sc; ssh[tid] = bn2b[tid] - mu * sc;
  }
  __syncthreads();
  for (int i = tid; i < 256 * 31; i += NT) {
    int c = i / 31;
    float v = sA[i] * ssc[c] + ssh[c];
    sA[i] = v > 0.f ? v : 0.f;
  }
  __syncthreads();
  // conv2: 256->128, 31->29
  for (int i = tid; i < 128 * 29; i += NT) {
    int c = i / 29, t = i % 29;
    float acc = e2b[c];
    const float* wb = e2w + c * 256 * 3;
    for (int ci = 0; ci < 256; ++ci) {
      const float* ip = sA + ci * 31 + t;
      const float* wp = wb + ci * 3;
      acc += ip[0] * wp[0] + ip[1] * wp[1] + ip[2] * wp[2];
    }
    sB[i] = acc;
  }
  __syncthreads();
  // bn3 + relu
  if (tid < 128) {
    float s = 0.f, q = 0.f;
    for (int t = 0; t < 29; ++t) { float v = sB[tid * 29 + t]; s += v; q += v * v; }
    float mu = s * (1.f / 29.f);
    float var = q * (1.f / 29.f) - mu * mu;
    float sc = bn3g[tid] * rsqrtf(var + BN_EPS);
    ssc[tid] = sc; ssh[tid] = bn3b[tid] - mu * sc;
  }
  __syncthreads();
  for (int i = tid; i < 128 * 29; i += NT) {
    int c = i / 29;
    float v = sB[i] * ssc[c] + ssh[c];
    sB[i] = v > 0.f ? v : 0.f;
  }
  __syncthreads();
  // conv3: 128->64, 29->27
  for (int i = tid; i < 64 * 27; i += NT) {
    int c = i / 27, t = i % 27;
    float acc = e3b[c];
    const float* wb = e3w + c * 128 * 3;
    for (int ci = 0; ci < 128; ++ci) {
      const float* ip = sB + ci * 29 + t;
      const float* wp = wb + ci * 3;
      acc += ip[0] * wp[0] + ip[1] * wp[1] + ip[2] * wp[2];
    }
    sC[i] = acc;
  }
  __syncthreads();
  // relu THEN bn4 (no relu after)
  for (int i = tid; i < 64 * 27; i += NT) { float v = sC[i]; sC[i] = v > 0.f ? v : 0.f; }
  __syncthreads();
  if (tid < 64) {
    float s = 0.f, q = 0.f;
    for (int t = 0; t < 27; ++t) { float v = sC[tid * 27 + t]; s += v; q += v * v; }
    float mu = s * (1.f / 27.f);
    float var = q * (1.f / 27.f) - mu * mu;
    float sc = bn4g[tid] * rsqrtf(var + BN_EPS);
    ssc[tid] = sc; ssh[tid] = bn4b[tid] - mu * sc;
  }
  __syncthreads();
  for (int i = tid; i < 64 * 27; i += NT) {
    int c = i / 27;
    sC[i] = sC[i] * ssc[c] + ssh[c];
  }
  __syncthreads();
  // conv4: 64->32, 27->25 -> H
  for (int i = tid; i < 32 * 25; i += NT) {
    int c = i / 25, t = i % 25;
    float acc = e4b[c];
    const float* wb = e4w + c * 64 * 3;
    for (int ci = 0; ci < 64; ++ci) {
      const float* ip = sC + ci * 27 + t;
      const float* wp = wb + ci * 3;
      acc += ip[0] * wp[0] + ip[1] * wp[1] + ip[2] * wp[2];
    }
    H[i] = acc;
  }
}

// ---------------------------------------------------------------------------
// K2: z = efc_w (32768x800) @ h + efc_b via V_WMMA_F32_16X16X4_F32.
// One wave = 16 rows. A layout (ISA 16x4 f32): lane m (0..15) row m,
// half-wave selects K pair {0,1}/{2,3}. B = h replicated across all N.
// h staged into LDS by the Tensor Data Mover when available.
// ---------------------------------------------------------------------------
__global__ __launch_bounds__(256) void k_efc(
    const float* __restrict__ W, const float* __restrict__ bias,
    const float* __restrict__ H, float* __restrict__ Z)
{
  __shared__ float sh[800];
  const int tid = threadIdx.x;
#if USE_TDM
  if (tid < 32) {                                  // wave 0 drives the TDM
    tdm_load_1d(H, lds_off(sh), 800, 800);
    __builtin_amdgcn_s_wait_tensorcnt(0);
  }
  __syncthreads();
#else
  for (int i = tid; i < 800; i += 256) sh[i] = H[i];
  __syncthreads();
#endif

  const int wave = tid >> 5;
  const int lane = tid & 31;
  const int row0 = (blockIdx.x * 8 + wave) * 16;
  const int m = lane & 15;
  const int koff = (lane >> 4) << 1;             // 0 or 2
  const v2f* __restrict__ wrow =
      (const v2f*)(W + (size_t)(row0 + m) * 800 + koff);   // 8B aligned

  v8f acc = {0.f, 0.f, 0.f, 0.f, 0.f, 0.f, 0.f, 0.f};
#pragma unroll 4
  for (int k = 0; k < 800; k += 4) {
    if ((k & 15) == 0)                           // 1 near-cache prefetch / 4 WMMA
      __builtin_prefetch((const float*)wrow + k + 256, 0, 3);
    v2f a = wrow[k >> 1];                        // floats [koff+k, koff+k+1]
    v2f b;
    b.x = sh[k + koff];
    b.y = sh[k + koff + 1];
    acc = __builtin_amdgcn_wmma_f32_16x16x4_f32(
        false, a, false, b, (short)0, acc, false, false);
  }
  // D layout: VGPR v -> row v (lanes 0-15) / row v+8 (lanes 16-31), N = lane%16.
  if (m == 0) {
    int base = row0 + ((lane >> 4) << 3);
#pragma unroll
    for (int r = 0; r < 8; ++r) Z[base + r] = acc[r] + bias[base + r];
  }
}

// ---------------------------------------------------------------------------
// K3: Af = dot(X,Y)/(dot(X,X)+0.01); P[f] = X0*Af^f; zero DFC accumulator.
// ---------------------------------------------------------------------------
__global__ __launch_bounds__(1024) void k_mid(
    const float* __restrict__ Z, float* __restrict__ P,
    float* __restrict__ SC, float* __restrict__ DFC)
{
  __shared__ float ra[1024], rb[1024];
  __shared__ float sAf;
  const int tid = threadIdx.x;
  float sxy = 0.f, sxx = 0.f;
  for (int i = tid; i < 32767; i += 1024) {
    float xv = Z[i];
    sxy += xv * Z[i + 1];
    sxx += xv * xv;
  }
  ra[tid] = sxy; rb[tid] = sxx;
  __syncthreads();
  for (int s = 512; s > 0; s >>= 1) {
    if (tid < s) { ra[tid] += ra[tid + s]; rb[tid] += rb[tid + s]; }
    __syncthreads();
  }
  if (tid == 0) {
    float Af = ra[0] / (rb[0] + 0.01f);
    sAf = Af; SC[0] = Af;
  }
  __syncthreads();
  const float Af = sAf;
  const float X0 = Z[0];
  const float la = __logf(fabsf(Af));
  const bool neg = (Af < 0.f);
  for (int f = tid; f < 32767; f += 1024) {
    float mag = (f == 0) ? 1.f : __expf(la * (float)f);
    float v = X0 * mag;
    if (neg && (f & 1)) v = -v;
    P[f] = v;
  }
  for (int i = tid; i < 1600; i += 1024) DFC[i] = 0.f;
}

// ---------------------------------------------------------------------------
// K5: DFC[2x800] += out2 (2x32767) @ dfc_w.T via WMMA.
// K split into 64 chunks of 512; X / powers chunks staged into LDS by TDM
// (OOB zero-fill covers the ragged tail). B col 0 = X, col 1 = P.
// Partials merged with global_atomic_add_f32.
// ---------------------------------------------------------------------------
__global__ __launch_bounds__(256) void k_dfc(
    const float* __restrict__ Wd, const float* __restrict__ Z,
    const float* __restrict__ P, float* __restrict__ DFC)
{
  __shared__ float sx[512], sp[512];
  const int tid = threadIdx.x;
  const int f0 = blockIdx.x * 512;
#if USE_TDM
  if (tid < 32) {                                  // wave 0 drives the TDM
    int rem = 32767 - f0;
    if (rem > 512) rem = 512;
    tdm_load_1d(Z + f0, lds_off(sx), rem, 512);    // OOB tail -> zeros
    tdm_load_1d(P + f0, lds_off(sp), rem, 512);
    __builtin_amdgcn_s_wait_tensorcnt(0);
  }
  __syncthreads();
#else
  for (int i = tid; i < 512; i += 256) {
    int f = f0 + i;
    bool ok = (f < 32767);
    sx[i] = ok ? Z[f] : 0.f;
    sp[i] = ok ? P[f] : 0.f;
  }
  __syncthreads();
#endif

  const int wave = tid >> 5;
  const int lane = tid & 31;
  const int tile = blockIdx.y * 8 + wave;
  if (tile >= 50) return;                        // whole-wave exit, post-barrier

  const int m = lane & 15;                       // A row within tile / B column
  const int kh = lane >> 4;
  const int koff = kh << 1;
  const int row = tile * 16 + m;
  const float* __restrict__ arow = Wd + (size_t)row * 32767;

  v8f acc = {0.f, 0.f, 0.f, 0.f, 0.f, 0.f, 0.f, 0.f};
#pragma unroll 4
  for (int k = 0; k < 512; k += 4) {
    int f = f0 + k + koff;
    v2f a;
    a.x = (f < 32767) ? arow[f] : 0.f;
    a.y = (f + 1 < 32767) ? arow[f + 1] : 0.f;
    float vx0 = sx[k + koff], vx1 = sx[k + koff + 1];
    float vp0 = sp[k + koff], vp1 = sp[k + koff + 1];
    v2f b;
    b.x = (m == 0) ? vx0 : ((m == 1) ? vp0 : 0.f);
    b.y = (m == 0) ? vx1 : ((m == 1) ? vp1 : 0.f);
    acc = __builtin_amdgcn_wmma_f32_16x16x4_f32(
        false, a, false, b, (short)0, acc, false, false);
  }
  // lanes (n=0,1) x (halves kh=0,1) carry the two useful D columns.
  if (m < 2) {
    const int jbase = tile * 16 + (kh << 3);
#pragma unroll
    for (int v = 0; v < 8; ++v) atomicAdd(&DFC[m * 800 + jbase + v], acc[v]);
  }
}

// ---------------------------------------------------------------------------
// K6: decoder — one workgroup, small global scratch buffers.
// ---------------------------------------------------------------------------
__device__ __forceinline__ void convT_dev(
    const float* __restrict__ in, int Cin, int Lin,
    const float* __restrict__ w, const float* __restrict__ bias,
    int Cout, int k, float* __restrict__ out, int tid, int NT)
{
  const int Lout = Lin + k - 1;
  for (int idx = tid; idx < Cout * Lout; idx += NT) {
    int o = idx / Lout, t = idx % Lout;
    float acc = bias[o];
    int s0 = t - (Lin - 1); if (s0 < 0) s0 = 0;
    int s1 = (t < k - 1) ? t : (k - 1);
    for (int i = 0; i < Cin; ++i) {
      const float* ip = in + i * Lin + t;
      const float* wp = w + (i * Cout + o) * k;
      for (int s = s0; s <= s1; ++s) acc += ip[-s] * wp[s];
    }
    out[idx] = acc;
  }
}

__device__ __forceinline__ void bn_stats_dev(
    const float* __restrict__ in, int C, int L,
    const float* __restrict__ g, const float* __restrict__ b,
    float* ssc, float* ssh, bool relu_in, int tid)
{
  if (tid < C) {
    const float* p = in + tid * L;
    float s = 0.f, q = 0.f;
    for (int t = 0; t < L; ++t) {
      float v = p[t];
      if (relu_in && v < 0.f) v = 0.f;
      s += v; q += v * v;
    }
    float mu = s / (float)L;
    float var = q / (float)L - mu * mu;
    float sc = g[tid] * rsqrtf(var + BN_EPS);
    ssc[tid] = sc; ssh[tid] = b[tid] - mu * sc;
  }
}

__device__ __forceinline__ void bn_apply_dev(
    const float* __restrict__ in, float* __restrict__ out, int C, int L,
    const float* ssc, const float* ssh, bool relu_in, bool relu_out,
    int tid, int NT)
{
  for (int i = tid; i < C * L; i += NT) {
    int c = i / L;
    float v = in[i];
    if (relu_in && v < 0.f) v = 0.f;
    v = v * ssc[c] + ssh[c];
    if (relu_out && v < 0.f) v = 0.f;
    out[i] = v;
  }
}

__global__ __launch_bounds__(1024) void k_decoder(
    const float* __restrict__ DFC, const float* __restrict__ dfc_b,
    const float* __restrict__ dbn4g, const float* __restrict__ dbn4b,
    const float* __restrict__ d4w, const float* __restrict__ d4b,
    const float* __restrict__ dbn3g, const float* __restrict__ dbn3b,
    const float* __restrict__ d3w, const float* __restrict__ d3b,
    const float* __restrict__ dbn2g, const float* __restrict__ dbn2b,
    const float* __restrict__ d2w, const float* __restrict__ d2b,
    const float* __restrict__ dbn1g, const float* __restrict__ dbn1b,
    const float* __restrict__ d1w, const float* __restrict__ d1b,
    float* __restrict__ B1, float* __restrict__ B2, float* __restrict__ T0,
    float* __restrict__ out)
{
  __shared__ float ssc[256], ssh[256];
  const int tid = threadIdx.x;
  const int NT = 1024;

  // d = dfc output + bias, flattened (2,800) -> (32,50)
  for (int i = tid; i < 1600; i += NT) T0[i] = DFC[i] + dfc_b[i % 800];
  __syncthreads();
  // relu(bn(d, dbn4)) -> B1
  bn_stats_dev(T0, 32, 50, dbn4g, dbn4b, ssc, ssh, false, tid);
  __syncthreads();
  bn_apply_dev(T0, B1, 32, 50, ssc, ssh, false, true, tid, NT);
  __syncthreads();
  // convT d4: 32->64, 50->54
  convT_dev(B1, 32, 50, d4w, d4b, 64, 5, B2, tid, NT);
  __syncthreads();
  // bn(relu(d), dbn3) -> B1
  bn_stats_dev(B2, 64, 54, dbn3g, dbn3b, ssc, ssh, true, tid);
  __syncthreads();
  bn_apply_dev(B2, B1, 64, 54, ssc, ssh, true, false, tid, NT);
  __syncthreads();
  // convT d3: 64->128, 54->58
  convT_dev(B1, 64, 54, d3w, d3b, 128, 5, B2, tid, NT);
  __syncthreads();
  // bn(relu(d), dbn2) -> B1
  bn_stats_dev(B2, 128, 58, dbn2g, dbn2b, ssc, ssh, true, tid);
  __syncthreads();
  bn_apply_dev(B2, B1, 128, 58, ssc, ssh, true, false, tid, NT);
  __syncthreads();
  // convT d2: 128->256, 58->62
  convT_dev(B1, 128, 58, d2w, d2b, 256, 5, B2, tid, NT);
  __syncthreads();
  // relu(bn(d, dbn1)) -> B1
  bn_stats_dev(B2, 256, 62, dbn1g, dbn1b, ssc, ssh, false, tid);
  __syncthreads();
  bn_apply_dev(B2, B1, 256, 62, ssc, ssh, false, true, tid, NT);
  __syncthreads();
  // convT d1: 256->1, 62->64 -> out
  convT_dev(B1, 256, 62, d1w, d1b, 1, 3, out, tid, NT);
}

// ---------------------------------------------------------------------------
extern "C" void kernel_launch(void* const* d_in, const int* in_sizes, int n_in,
                              void* d_out, int out_size, void* d_ws, size_t ws_size,
                              hipStream_t stream)
{
  const float* x     = (const float*)d_in[0];
  const float* e1w   = (const float*)d_in[1];
  const float* e1b   = (const float*)d_in[2];
  const float* bn2g  = (const float*)d_in[3];
  const float* bn2b  = (const float*)d_in[4];
  const float* e2w   = (const float*)d_in[5];
  const float* e2b   = (const float*)d_in[6];
  const float* bn3g  = (const float*)d_in[7];
  const float* bn3b  = (const float*)d_in[8];
  const float* e3w   = (const float*)d_in[9];
  const float* e3b   = (const float*)d_in[10];
  const float* bn4g  = (const float*)d_in[11];
  const float* bn4b  = (const float*)d_in[12];
  const float* e4w   = (const float*)d_in[13];
  const float* e4b   = (const float*)d_in[14];
  const float* efc_w = (const float*)d_in[15];
  const float* efc_b = (const float*)d_in[16];
  const float* dfc_w = (const float*)d_in[17];
  const float* dfc_b = (const float*)d_in[18];
  const float* dbn4g = (const float*)d_in[19];
  const float* dbn4b = (const float*)d_in[20];
  const float* d4w   = (const float*)d_in[21];
  const float* d4b   = (const float*)d_in[22];
  const float* dbn3g = (const float*)d_in[23];
  const float* dbn3b = (const float*)d_in[24];
  const float* d3w   = (const float*)d_in[25];
  const float* d3b   = (const float*)d_in[26];
  const float* dbn2g = (const float*)d_in[27];
  const float* dbn2b = (const float*)d_in[28];
  const float* d2w   = (const float*)d_in[29];
  const float* d2b   = (const float*)d_in[30];
  const float* dbn1g = (const float*)d_in[31];
  const float* dbn1b = (const float*)d_in[32];
  const float* d1w   = (const float*)d_in[33];
  const float* d1b   = (const float*)d_in[34];

  float* ws  = (float*)d_ws;
  float* H   = ws + 0;       // 800
  float* Z   = ws + 1024;    // 32768
  float* P   = ws + 34816;   // 32767
  float* SCL = ws + 67600;   // scalars
  float* DFC = ws + 67712;   // 1600 (atomic accumulator)
  float* T0  = ws + 69376;   // 1600
  float* B1  = ws + 71680;   // 16384
  float* B2  = ws + 88064;   // 16384

  k_encoder<<<1, 1024, 0, stream>>>(x, e1w, e1b, bn2g, bn2b, e2w, e2b,
                                    bn3g, bn3b, e3w, e3b, bn4g, bn4b,
                                    e4w, e4b, H);
  k_efc<<<256, 256, 0, stream>>>(efc_w, efc_b, H, Z);
  k_mid<<<1, 1024, 0, stream>>>(Z, P, SCL, DFC);
  k_dfc<<<dim3(64, 7, 1), 256, 0, stream>>>(dfc_w, Z, P, DFC);
  k_decoder<<<1, 1024, 0, stream>>>(DFC, dfc_b, dbn4g, dbn4b, d4w, d4b,
                                    dbn3g, dbn3b, d3w, d3b, dbn2g, dbn2b,
                                    d2w, d2b, dbn1g, dbn1b, d1w, d1b,
                                    B1, B2, T0, (float*)d_out);
}